// Q_1In1Out_Conv1D_6966436954607
// MI455X (gfx1250) — compile-verified
//
#include <hip/hip_runtime.h>
#include <stdint.h>

typedef float v2f __attribute__((ext_vector_type(2)));
typedef float v4f __attribute__((ext_vector_type(4)));
typedef float v8f __attribute__((ext_vector_type(8)));
typedef unsigned int v4u __attribute__((ext_vector_type(4)));
typedef int v4i __attribute__((ext_vector_type(4)));
typedef int v8i __attribute__((ext_vector_type(8)));

#define ROWLEN 8192
#define NPATCH 8188            // ROWLEN - 5 + 1   (== 4 mod 8; see store logic)
#define TILE_P 512             // patches per block
#define TILE_E (TILE_P + 8)    // staged elements (512 + 4 halo, padded)
#define ITERS  4               // 8 waves * 4 iters * 16 patches = 512

// z(p) = cos(th)*cA1*C + sin(th)*sA0*cB0*sA1*( sin(ph)*sB1*C - cos(ph)*cB1 )
//   cA = 1/sqrt(1+x^2), sA = x/sqrt(1+x^2), cB = 1/sqrt(1+x^4), sB = x^2/sqrt(1+x^4)
//   C  = cA2*cA3*cA4   (indices are window offsets 0..4)

__global__ __launch_bounds__(256) void qconv_closed_form(const float* __restrict__ x,
                                                         const float* __restrict__ w,
                                                         float* __restrict__ out) {
    __shared__ float sm[TILE_E];

    const int tid  = threadIdx.x;
    const int lane = tid & 31;
    const int wave = tid >> 5;
    const int half = lane >> 4;     // 0: supplies K=0,1 of A; 1: supplies K=2,3
    const int m    = lane & 15;     // A-matrix row = patch within group of 16

    const int row   = blockIdx.y;
    const int tile0 = blockIdx.x * TILE_P;

    // ---- Stage x[row, tile0 .. tile0+TILE_E) into LDS via the Tensor Data Mover.
    //      OOB (past row end) reads are zero-filled by TDM via tensor_dim0.
#if __has_builtin(__builtin_amdgcn_tensor_load_to_lds)
    if (wave == 0) {
        const uint64_t ga  = (uint64_t)(uintptr_t)(const void*)(x + (size_t)row * ROWLEN + tile0);
        const uint32_t lds = (uint32_t)(uintptr_t)(void*)sm;
        v4u g0;
        g0.x = 1u;                                               // count=1, user mode
        g0.y = lds;                                              // lds_addr
        g0.z = (uint32_t)ga;                                     // global_addr[31:0]
        g0.w = (uint32_t)((ga >> 32) & 0x01FFFFFFu) | 0x80000000u; // addr[56:32] | type=2
        const uint32_t dim0 = (uint32_t)(ROWLEN - tile0);        // elements left in row
        v8i g1;
        g1[0] = 0x00020000;                                      // data_size = 4 bytes
        g1[1] = (int)((dim0 & 0xFFFFu) << 16);                   // tensor_dim0[15:0]
        g1[2] = (int)((dim0 >> 16) | (1u << 16));                // tensor_dim0[31:16], tensor_dim1=1
        g1[3] = (int)((uint32_t)TILE_E << 16);                   // tile_dim0 = TILE_E
        g1[4] = 1;                                               // tile_dim1 = 1, tile_dim2 = 0
        g1[5] = ROWLEN;                                          // tensor_dim0_stride
        g1[6] = 0;
        g1[7] = 0;
        v4i gz = {0, 0, 0, 0};
#if defined(__clang_major__) && (__clang_major__ >= 23)
        v8i gz8 = {0, 0, 0, 0, 0, 0, 0, 0};
        __builtin_amdgcn_tensor_load_to_lds(g0, g1, gz, gz, gz8, 0);
#else
        __builtin_amdgcn_tensor_load_to_lds(g0, g1, gz, gz, 0);
#endif
        __builtin_amdgcn_s_wait_tensorcnt(0);
    }
#else
    // Fallback staging (should not be hit on this toolchain)
    for (int e = tid; e < TILE_E; e += 256) {
        int ge = tile0 + e;
        sm[e] = (ge < ROWLEN) ? x[(size_t)row * ROWLEN + ge] : 0.0f;
    }
#endif
    __syncthreads();

    // ---- Weight-dependent coefficients (omega is provably irrelevant).
    const float ph = w[0], th = w[1];
    const float ct = cosf(th), st = sinf(th);
    const float cp = cosf(ph), sp = sinf(ph);

    // B matrix (4x16, columns identical): B0=cos(th), B1=-sin(th)cos(ph), B2=sin(th)sin(ph), B3=0
    v2f bv;
    bv.x = half ? st * sp : ct;
    bv.y = half ? 0.0f   : -(st * cp);

    float* orow = out + (size_t)row * NPATCH;

    for (int it = 0; it < ITERS; ++it) {
        const int pb = wave * (ITERS * 16) + it * 16;   // first patch of this 16-group
        const int pl = pb + m;                          // this lane's patch (local)

        const float x0 = sm[pl + 0];
        const float x1 = sm[pl + 1];
        const float x2 = sm[pl + 2];
        const float x3 = sm[pl + 3];
        const float x4 = sm[pl + 4];

        const float r0 = rsqrtf(fmaf(x0, x0, 1.0f));
        const float r1 = rsqrtf(fmaf(x1, x1, 1.0f));
        const float r2 = rsqrtf(fmaf(x2, x2, 1.0f));
        const float r3 = rsqrtf(fmaf(x3, x3, 1.0f));
        const float r4 = rsqrtf(fmaf(x4, x4, 1.0f));
        const float x0_2 = x0 * x0, x1_2 = x1 * x1;
        const float q0 = rsqrtf(fmaf(x0_2, x0_2, 1.0f));
        const float q1 = rsqrtf(fmaf(x1_2, x1_2, 1.0f));

        const float C  = r2 * r3 * r4;                  // cA2*cA3*cA4
        const float G  = (x0 * r0) * q0 * (x1 * r1);    // sA0*cB0*sA1
        const float A0 = r1 * C;                        // cA1*C
        const float A1 = G * q1;                        // G*cB1
        const float A2 = G * (x1_2 * q1) * C;           // G*sB1*C

        // A matrix (16x4): lanes 0-15 carry K=0,1; lanes 16-31 carry K=2,3 of the same row.
        v2f av;
        av.x = half ? A2 : A0;
        av.y = half ? 0.0f : A1;

        v8f acc = {0.f, 0.f, 0.f, 0.f, 0.f, 0.f, 0.f, 0.f};
        v8f d = __builtin_amdgcn_wmma_f32_16x16x4_f32(
            /*neg_a=*/false, av, /*neg_b=*/false, bv,
            /*c_mod=*/(short)0, acc, /*reuse_a=*/false, /*reuse_b=*/false);

        // D[m,n] identical across columns n. Lane 0 holds M=0..7 in d[0..7],
        // lane 16 holds M=8..15 in d[0..7].
        //
        // NPATCH == 4 (mod 8) and mb is 8-aligned, so:
        //   - the low float4 (mb..mb+3) is ALWAYS in bounds (max mb = 8184 -> 8187 = NPATCH-1)
        //   - the high float4 is in bounds iff mb+8 <= NPATCH (false only for the very last group)
        // => no scalar tail needed, no OOB writes possible.
        if (m == 0) {
            const int mb = tile0 + pb + half * 8;       // 8-aligned; orow+mb is 16B aligned
            v4f lo = {d[0], d[1], d[2], d[3]};
            v4f hi = {d[4], d[5], d[6], d[7]};
            *reinterpret_cast<v4f*>(orow + mb) = lo;            // global_store_b128 (always)
            if (mb + 8 <= NPATCH) {
                *reinterpret_cast<v4f*>(orow + mb + 4) = hi;    // global_store_b128 (predicated)
            }
        }
    }
}

extern "C" void kernel_launch(void* const* d_in, const int* in_sizes, int n_in,
                              void* d_out, int out_size, void* d_ws, size_t ws_size,
                              hipStream_t stream) {
    const float* x   = (const float*)d_in[0];   // (B, 8192) f32
    const float* wts = (const float*)d_in[1];   // (3,) f32
    float* out = (float*)d_out;                 // (B, 8188) f32

    const int B = in_sizes[0] / ROWLEN;
    dim3 grid((NPATCH + TILE_P - 1) / TILE_P, B);  // (16, 64)
    dim3 block(256);
    hipLaunchKernelGGL(qconv_closed_form, grid, block, 0, stream, x, wts, out);
}